// NeRFRenderer_37125697306879
// MI455X (gfx1250) — compile-verified
//
#include <hip/hip_runtime.h>
#include <math.h>

#define N_RAYS     2048
#define TOT        128
#define IND_S      16
#define HID        64
#define MAX_DEPTHS 2
#define EPSF       1e-8f
#define M_SAMPLES  (N_RAYS * TOT)
#define PI_F       3.14159265358979323846f

typedef float v2f __attribute__((ext_vector_type(2)));
typedef float v8f __attribute__((ext_vector_type(8)));

// Fast transcendentals: inputs here are bounded (MLP pre-activations, optical
// depths), so the hardware v_sin_f32 / v_exp_f32 paths are plenty accurate and
// avoid the Payne-Hanek / ldexp slow paths that bloated the previous build.
__device__ __forceinline__ float softplus_f(float x) {
  return fmaxf(x, 0.0f) + log1pf(__expf(-fabsf(x)));
}
__device__ __forceinline__ float sigmoid_f(float x) {
  float xc = fminf(fmaxf(x, -30.0f), 30.0f);
  return 1.0f / (1.0f + __expf(-xc));
}
__device__ __forceinline__ float tanh_f(float x) {
  float xc = fminf(fmaxf(x, -15.0f), 15.0f);
  float e  = __expf(2.0f * xc);
  return (e - 1.0f) / (e + 1.0f);
}

// One bilinear tap of the 3x128x256 environment map (matches reference tap()).
__device__ __forceinline__ void env_tap(const float* __restrict__ env,
                                        float xc, float yc, float w,
                                        float& v0, float& v1, float& v2) {
  bool inb = (xc >= 0.0f) && (xc < 256.0f) && (yc >= 0.0f) && (yc < 128.0f);
  float wm = inb ? w : 0.0f;
  int xi = (int)fminf(fmaxf(xc, 0.0f), 255.0f);
  int yi = (int)fminf(fmaxf(yc, 0.0f), 127.0f);
  int base = yi * 256 + xi;
  v0 += wm * env[base];
  v1 += wm * env[32768 + base];
  v2 += wm * env[65536 + base];
}

__global__ __launch_bounds__(256)
void nerf_render_kernel(const float* __restrict__ rays_o,
                        const float* __restrict__ rays_d,
                        const float* __restrict__ env,
                        const float* __restrict__ Wd1,   // (3,64)
                        const float* __restrict__ Wd2,   // (64,3)
                        const float* __restrict__ Wf1,   // (6,64)
                        const float* __restrict__ Ws,    // (64,3)
                        const float* __restrict__ Wdir,  // (64,3)
                        const float* __restrict__ Wrho,  // (64,3)
                        const float* __restrict__ jitter,    // (2,M,16)
                        const float* __restrict__ u_t,       // (2,M)
                        const float* __restrict__ u_scatter, // (2,M)
                        const int*   __restrict__ channel,   // (2,M)
                        float* __restrict__ out)             // (2048,3)
{
  // Block = 256 threads (8 wave32) = 256 samples = 2 full rays.
  __shared__ float  sState[256][9];   // o(3), d(3), nears, span  (pad to 9)
  __shared__ float  sMax[256][3];     // per-sample pre-clamp channel majorants
  __shared__ float  sH[8][16][68];    // per-wave hidden transpose buffer (bank-clean stride)
  __shared__ float2 sB2[16][32];      // Wd2 in WMMA B-layout, one shared copy (saves 32 VGPRs)
  __shared__ float  sWf1[6 * 64];
  __shared__ float  sWs[64 * 3];
  __shared__ float  sWdir[64 * 3];
  __shared__ float  sWrho[64 * 3];
  __shared__ float  sRed[256][3];

  const int tid  = threadIdx.x;
  const int lane = tid & 31;
  const int wave = tid >> 5;
  const int hi   = (lane >> 4) & 1;   // lane half (K-split of WMMA operands)
  const int lm   = lane & 15;
  const int g    = blockIdx.x * 256 + tid;   // global sample index

  // Stage small MLP weights to LDS.
  for (int k = tid; k < 6 * 64; k += 256) sWf1[k] = Wf1[k];
  for (int k = tid; k < 192; k += 256) {
    sWs[k] = Ws[k]; sWdir[k] = Wdir[k]; sWrho[k] = Wrho[k];
  }

  // Layer-1 B operand (Wd1, K=4 padded x N=16 per tile) stays in 8 VGPRs:
  // B layout: VGPR0 = K0(lanes0-15)/K2(lanes16-31), VGPR1 = K1/K3 (K=3 zero pad).
  v2f b1[4];
#pragma unroll
  for (int t = 0; t < 4; ++t) {
    int n = lm + 16 * t;
    b1[t].x = Wd1[(hi ? 2 : 0) * 64 + n];
    b1[t].y = hi ? 0.0f : Wd1[64 + n];
  }
  // Layer-2 B operand (Wd2, 16 K-chunks of K=4 x N=16, only N<3 valid) goes to
  // LDS in B-layout; identical per wave, so wave 0 writes it once.
  if (wave == 0) {
#pragma unroll
    for (int k = 0; k < 16; ++k) {
      int k0 = 4 * k + 2 * hi;
      float bx = 0.0f, by = 0.0f;
      if (lm < 3) { bx = Wd2[k0 * 3 + lm]; by = Wd2[(k0 + 1) * 3 + lm]; }
      sB2[k][lane] = make_float2(bx, by);
    }
  }

  // Per-thread persistent ray state.
  const int ray = g / TOT;
  float ox = rays_o[ray * 3 + 0], oy = rays_o[ray * 3 + 1], oz = rays_o[ray * 3 + 2];
  float dx = rays_d[ray * 3 + 0], dy = rays_d[ray * 3 + 1], dz = rays_d[ray * 3 + 2];
  float th0 = 1.0f, th1 = 1.0f, th2 = 1.0f;     // thr
  float rg0 = 0.0f, rg1 = 0.0f, rg2 = 0.0f;     // rgbs
  bool  alive = true;
  float nears = 0.0f, fars = 0.0f, span = 0.0f;

  __syncthreads();

  for (int dep = 0; dep < MAX_DEPTHS; ++dep) {
    // ---------------- Phase 1: per-thread ray/sphere intersection ----------------
    {
      float aq = dx * dx + dy * dy + dz * dz;
      float bq = 2.0f * (dx * ox + dy * oy + dz * oz);
      float cq = ox * ox + oy * oy + oz * oz - 1.0f;  // BOUND^2 = 1
      float delta = bq * bq - 4.0f * aq * cq;
      alive = alive && (delta > 0.0f);
      float sq = sqrtf(fmaxf(delta, 0.0f));
      float inv2a = 0.5f / aq;
      nears = fmaxf((-bq - sq) * inv2a, 0.001f);
      fars  = fmaxf((-bq + sq) * inv2a, 0.001f);
      span  = fars - nears;
      sState[tid][0] = ox; sState[tid][1] = oy; sState[tid][2] = oz;
      sState[tid][3] = dx; sState[tid][4] = dy; sState[tid][5] = dz;
      sState[tid][6] = nears; sState[tid][7] = span;
    }
    __syncthreads();

    // ---------------- Phase 2: density MLP on the matrix pipe ----------------
    // One sample's 16 jittered points = one 16-row WMMA tile; each wave handles
    // its own 32 samples. Keep the loop rolled to bound register pressure.
#pragma unroll 1
    for (int it = 0; it < 32; ++it) {
      const int sl = wave * 32 + it;             // block-local sample
      const int gs = blockIdx.x * 256 + sl;      // global sample
      float sox = sState[sl][0], soy = sState[sl][1], soz = sState[sl][2];
      float sdx = sState[sl][3], sdy = sState[sl][4], sdz = sState[sl][5];
      float snr = sState[sl][6], ssp = sState[sl][7];

      // Point for this lane's row (m = lane&15); both lane halves need full xyz.
      float jt = jitter[(dep * M_SAMPLES + gs) * IND_S + lm];
      float z  = snr + ssp * ((float)lm * (1.0f / 15.0f))
                     + (jt - 0.5f) * ssp * (1.0f / 16.0f);
      float px = sox + sdx * z, py = soy + sdy * z, pz = soz + sdz * z;

      // A-layout for 16x16x4 f32: lanes 0-15 -> {K0=x, K1=y}, lanes 16-31 -> {K2=z, K3=0}.
      v2f amat; amat.x = hi ? pz : px; amat.y = hi ? 0.0f : py;

      const v8f zc = {0.0f, 0.0f, 0.0f, 0.0f, 0.0f, 0.0f, 0.0f, 0.0f};
      // Layer 1 tile-by-tile: wmma -> sin -> LDS store keeps one v8f live.
#pragma unroll
      for (int t = 0; t < 4; ++t) {
        v8f h = __builtin_amdgcn_wmma_f32_16x16x4_f32(
            false, amat, false, b1[t], (short)0, zc, false, false);
#pragma unroll
        for (int r = 0; r < 8; ++r)
          sH[wave][r + 8 * hi][lm + 16 * t] = __sinf(h[r]);
      }
      // Wave-local ordering: DS ops are in-order per wave; wait + compiler fence.
      asm volatile("s_wait_dscnt 0x0" ::: "memory");

      // Layer 2: 16 chained K=4 WMMAs; A (transposed hidden) and B (Wd2) both
      // streamed from LDS so only ~12 VGPRs stay live across the chain.
      v8f s2 = {0.0f, 0.0f, 0.0f, 0.0f, 0.0f, 0.0f, 0.0f, 0.0f};
#pragma unroll
      for (int k = 0; k < 16; ++k) {
        v2f a2;
        a2.x = sH[wave][lm][4 * k + 2 * hi];
        a2.y = sH[wave][lm][4 * k + 2 * hi + 1];
        float2 bt = sB2[k][lane];
        v2f b2; b2.x = bt.x; b2.y = bt.y;
        s2 = __builtin_amdgcn_wmma_f32_16x16x4_f32(
            false, a2, false, b2, (short)0, s2, false, false);
      }

      // Max over the 16 points: 8 rows in-register + cross-half shuffle.
      // softplus is monotonic -> max(softplus(x)) = softplus(max(x)).
      float mx = s2[0];
#pragma unroll
      for (int r = 1; r < 8; ++r) mx = fmaxf(mx, s2[r]);
      mx = fmaxf(mx, __shfl_xor(mx, 16, 32));
      if (lane < 3) sMax[sl][lane] = softplus_f(mx);
    }
    __syncthreads();

    // ---------------- Phase 3: per-thread transport / env / material ----------------
    {
      float mm0 = fmaxf(sMax[tid][0], 0.001f);
      float mm1 = fmaxf(sMax[tid][1], 0.001f);
      float mm2 = fmaxf(sMax[tid][2], 0.001f);
      int   ch  = channel[dep * M_SAMPLES + g];
      float maj = (ch == 0) ? mm0 : ((ch == 1) ? mm1 : mm2);

      float ts;
      if (dep < MAX_DEPTHS - 1)
        ts = -log1pf(-u_t[dep * M_SAMPLES + g]) / maj + nears;
      else
        ts = fars;
      bool hit = alive && (ts >= fars);

      float mmax = fmaxf(mm0, fmaxf(mm1, mm2));
      float dnm0 = __expf(-mm0 * span) / (mmax + EPSF) + EPSF;
      float dnm1 = __expf(-mm1 * span) / (mmax + EPSF) + EPSF;
      float dnm2 = __expf(-mm2 * span) / (mmax + EPSF) + EPSF;
      float dmean = (dnm0 + dnm1 + dnm2) * (1.0f / 3.0f);

      // Environment sample at p = o + d*fars.
      float p0 = ox + dx * fars, p1 = oy + dy * fars, p2 = oz + dz * fars;
      float tha = atan2f(p0, -p2); if (isnan(tha)) tha = 0.0f;
      tha *= (1.0f / PI_F);
      float ac = acosf(p1); if (isnan(ac)) ac = 0.0f;
      float ph = (2.0f / PI_F) * ac - 1.0f;
      float ix = ((tha + 1.0f) * 256.0f - 1.0f) * 0.5f;
      float iy = ((ph  + 1.0f) * 128.0f - 1.0f) * 0.5f;
      float x0f = floorf(ix), y0f = floorf(iy);
      float wx = ix - x0f, wy = iy - y0f;
      float v0 = 0.0f, v1 = 0.0f, v2 = 0.0f;
      env_tap(env, x0f,        y0f,        (1.0f - wx) * (1.0f - wy), v0, v1, v2);
      env_tap(env, x0f + 1.0f, y0f,        wx * (1.0f - wy),          v0, v1, v2);
      env_tap(env, x0f,        y0f + 1.0f, (1.0f - wx) * wy,          v0, v1, v2);
      env_tap(env, x0f + 1.0f, y0f + 1.0f, wx * wy,                   v0, v1, v2);
      float le0 = __expf(v0), le1 = __expf(v1), le2 = __expf(v2);

      if (hit) {
        rg0 += th0 * (dnm0 / dmean) * le0;
        rg1 += th1 * (dnm1 / dmean) * le1;
        rg2 += th2 * (dnm2 / dmean) * le2;
      }
      alive = alive && !hit;

      if (dep < MAX_DEPTHS - 1) {
        float tr0 = __expf(-mm0 * (ts - nears)) / mmax;
        float tr1 = __expf(-mm1 * (ts - nears)) / mmax;
        float tr2 = __expf(-mm2 * (ts - nears)) / mmax;
        float dn2 = (mm0 * tr0 + mm1 * tr1 + mm2 * tr2) * (1.0f / 3.0f);
        if (alive) {
          float inv = 1.0f / dn2;
          th0 *= tr0 * inv; th1 *= tr1 * inv; th2 *= tr2 * inv;
          ox += dx * ts; oy += dy * ts; oz += dz * ts;
        }

        // Material MLP (6 -> 64 -> 3x3 heads), thread-per-sample fp32 VALU.
        float st0 = 0, st1 = 0, st2 = 0;
        float dd0 = 0, dd1 = 0, dd2 = 0;
        float rh0 = 0, rh1 = 0, rh2 = 0;
#pragma unroll 1
        for (int j = 0; j < HID; ++j) {
          float acc = ox * sWf1[j] + oy * sWf1[64 + j] + oz * sWf1[128 + j]
                    + dx * sWf1[192 + j] + dy * sWf1[256 + j] + dz * sWf1[320 + j];
          float f = tanh_f(acc);
          st0 += f * sWs[j * 3];     st1 += f * sWs[j * 3 + 1];     st2 += f * sWs[j * 3 + 2];
          dd0 += f * sWdir[j * 3];   dd1 += f * sWdir[j * 3 + 1];   dd2 += f * sWdir[j * 3 + 2];
          rh0 += f * sWrho[j * 3];   rh1 += f * sWrho[j * 3 + 1];   rh2 += f * sWrho[j * 3 + 2];
        }
        float sg0 = softplus_f(st0), sg1 = softplus_f(st1), sg2 = softplus_f(st2);
        float nrm = sqrtf(dd0 * dd0 + dd1 * dd1 + dd2 * dd2) + EPSF;
        float do0 = dd0 / nrm, do1 = dd1 / nrm, do2 = dd2 / nrm;
        float fr0 = sigmoid_f(rh0), fr1 = sigmoid_f(rh1), fr2 = sigmoid_f(rh2);

        float stc = (ch == 0) ? sg0 : ((ch == 1) ? sg1 : sg2);
        float sp  = fminf(stc / maj, 1.0f);
        bool  sm  = alive && (u_scatter[dep * M_SAMPLES + g] < sp);
        if (sm) { dx = do0; dy = do1; dz = do2; }
        if (sm) {
          float f = 1.0f / (sp + EPSF);
          th0 *= f * fr0; th1 *= f * fr1; th2 *= f * fr2;
        } else if (alive) {
          float f = 1.0f / (1.0f - sp + EPSF);
          th0 *= f; th1 *= f; th2 *= f;
        }
      }
    }
    __syncthreads();
  }

  // ---------------- Final: mean over the 128 samples of each of this block's 2 rays ----------------
  sRed[tid][0] = rg0; sRed[tid][1] = rg1; sRed[tid][2] = rg2;
  __syncthreads();
  if (tid < 6) {
    int r = tid / 3, c = tid % 3;
    float sum = 0.0f;
#pragma unroll 1
    for (int i = 0; i < 128; ++i) sum += sRed[r * 128 + i][c];
    out[(blockIdx.x * 2 + r) * 3 + c] = sum * (1.0f / 128.0f);
  }
}

extern "C" void kernel_launch(void* const* d_in, const int* in_sizes, int n_in,
                              void* d_out, int out_size, void* d_ws, size_t ws_size,
                              hipStream_t stream) {
  (void)in_sizes; (void)n_in; (void)out_size; (void)d_ws; (void)ws_size;
  const float* rays_o    = (const float*)d_in[0];
  const float* rays_d    = (const float*)d_in[1];
  const float* env_map   = (const float*)d_in[2];
  const float* Wd1       = (const float*)d_in[3];
  const float* Wd2       = (const float*)d_in[4];
  const float* Wf1       = (const float*)d_in[5];
  const float* Ws        = (const float*)d_in[6];
  const float* Wdir      = (const float*)d_in[7];
  const float* Wrho      = (const float*)d_in[8];
  const float* jitter    = (const float*)d_in[9];
  const float* u_t       = (const float*)d_in[10];
  const float* u_scatter = (const float*)d_in[11];
  const int*   channel   = (const int*)d_in[12];
  float* out = (float*)d_out;

  dim3 grid(M_SAMPLES / 256);
  dim3 block(256);
  nerf_render_kernel<<<grid, block, 0, stream>>>(
      rays_o, rays_d, env_map, Wd1, Wd2, Wf1, Ws, Wdir, Wrho,
      jitter, u_t, u_scatter, channel, out);
}